// TopKRouter_75282186764334
// MI455X (gfx1250) — compile-verified
//
#include <hip/hip_runtime.h>
#include <math.h>

// ---------------------------------------------------------------------------
// MI455X plan:
//   Kernel 1 (bandwidth):  global avg-pool of 64x512x56x56 f32 = 411MB read.
//       Roofline: 411MB / 23.3 TB/s ~= 17.6us. One wave32 per (b,c) row,
//       float4 non-temporal loads (read-once stream > 192MB L2), shfl reduce.
//       One EXTRA block computes the L2 regularizer concurrently (free).
//   Kernel 2 (latency):    fused MLP+router on ONE workgroup (8 waves),
//       all GEMMs via v_wmma_f32_16x16x4_f32 (full f32 precision),
//       activations ping-pong in LDS (stride 132 -> conflict-free A frags),
//       weights stream from global (L2 resident). Layers are templated on
//       <K,Mtiles,Ntiles,NVALID,RELU,TO_LDS> so the epilogue is branch-free
//       and bias is loaded once per tile.
// ---------------------------------------------------------------------------

typedef __attribute__((ext_vector_type(2))) float v2f;
typedef __attribute__((ext_vector_type(4))) float v4f;
typedef __attribute__((ext_vector_type(8))) float v8f;

#define BATCH 64
#define CHAN  512
#define HW    3136          // 56*56
#define POOL_ROWS (BATCH*CHAN)       // 32768
#define POOL_BLOCKS (POOL_ROWS/8)    // 8 waves/block -> 4096 blocks
#define LDH 132                      // LDS activation stride (bank-conflict pad)

struct Params12 { const float* p[12]; };

__device__ __forceinline__ float wave_reduce_add(float v) {
#pragma unroll
  for (int off = 16; off > 0; off >>= 1)
    v += __shfl_xor(v, off, 32);
  return v;
}

// ---------------------------------------------------------------------------
// Kernel 1: pooling (blocks 0..4095) + L2 regularizer (block 4096)
// ---------------------------------------------------------------------------
__global__ __launch_bounds__(256) void pool_l2_kernel(
    const float* __restrict__ in, float* __restrict__ pooled,
    float* __restrict__ l2_out, Params12 prm)
{
  if (blockIdx.x == POOL_BLOCKS) {
    // ---- L2 regularizer: sum_t sqrt(sum p_t^2), hidden under pooling ----
    __shared__ float red[8];
    const int l2n[12] = {128*512,128,128*128,128,64*128,64,64*64,64,64*64,64,1000*64,1000};
    float acc = 0.0f;
    for (int t = 0; t < 12; ++t) {
      const float* p = prm.p[t];
      const int n = l2n[t];
      float s = 0.0f;
      for (int i = threadIdx.x; i < n; i += 256) { float x = p[i]; s += x * x; }
      s = wave_reduce_add(s);
      if ((threadIdx.x & 31) == 0) red[threadIdx.x >> 5] = s;
      __syncthreads();
      if (threadIdx.x == 0) {
        float tot = 0.0f;
#pragma unroll
        for (int w = 0; w < 8; ++w) tot += red[w];
        acc += sqrtf(tot);
      }
      __syncthreads();
    }
    if (threadIdx.x == 0) *l2_out = 0.01f * acc;   // L2_LAMBDA
    return;
  }

  // ---- pooling: one wave32 per (b,c) row of 3136 contiguous floats ----
  const int lane = threadIdx.x & 31;
  const int row  = blockIdx.x * 8 + (threadIdx.x >> 5);
  const float* __restrict__ src = in + (size_t)row * HW;
  float s = 0.0f;
  for (int i = lane * 4; i < HW; i += 128) {
    v4f v = __builtin_nontemporal_load((const v4f*)(src + i));  // b128, TH=NT
    s += v.x + v.y + v.z + v.w;
  }
  s = wave_reduce_add(s);
  if (lane == 0) pooled[row] = s * (1.0f / (float)HW);
}

// ---------------------------------------------------------------------------
// One dense layer via V_WMMA_F32_16X16X4_F32 (compile-time shape).
// A: [64 x K] (lda stride), row-major (global or LDS). W: [N x K] row-major
// (computes A @ W^T + bias). Output to LDS (TO_LDS=1) or global (TO_LDS=0).
// A frag (16x4 f32): lane L -> m = mt*16+(L&15), k = k0 + 2*(L>>4) + v
// B frag (4x16 f32): lane L -> n = nt*16+(L&15), k = k0 + 2*(L>>4) + v
// C/D (16x16 f32):   vgpr r, lane L -> m = mt*16 + r + 8*(L>>4), n = nt*16+(L&15)
// ---------------------------------------------------------------------------
template <int K, int Mtiles, int Ntiles, int NVALID, bool RELU, bool TO_LDS>
__device__ __forceinline__ void wmma_layer(
    const float* __restrict__ A, int lda,
    const float* __restrict__ W,
    const float* __restrict__ bias,
    float* __restrict__ outp, int ldout)
{
  constexpr bool FULL = (NVALID == Ntiles * 16);   // no column guard needed
  constexpr int  NTILES = Mtiles * Ntiles;
  const int lane = threadIdx.x & 31;
  const int wave = threadIdx.x >> 5;
  const int idx  = lane & 15;
  const int hi   = lane >> 4;

  for (int t = wave; t < NTILES; t += 8) {          // uniform per wave
    const int mt = t % Mtiles;
    const int nt = t / Mtiles;
    const int m  = mt * 16 + idx;
    const int n  = nt * 16 + idx;
    const float* arow = A + (size_t)m * lda + 2 * hi;
    const float* brow = W + (size_t)n * K   + 2 * hi;
    const bool nval = FULL || (n < NVALID);
    const float bv = nval ? bias[n] : 0.0f;         // one bias load per tile
    v8f c = {};
#pragma unroll 16
    for (int k0 = 0; k0 < K; k0 += 4) {
      v2f a = *(const v2f*)(arow + k0);
      v2f b = {0.0f, 0.0f};
      if (nval) b = *(const v2f*)(brow + k0);
      c = __builtin_amdgcn_wmma_f32_16x16x4_f32(
              /*neg_a=*/false, a, /*neg_b=*/false, b,
              /*c_mod=*/(short)0, c, /*reuse_a=*/false, /*reuse_b=*/false);
    }
#pragma unroll
    for (int r = 0; r < 8; ++r) {
      const int mm = mt * 16 + r + 8 * hi;
      if (nval) {
        float v = c[r] + bv;
        if (RELU) v = fmaxf(v, 0.0f);
        outp[(size_t)mm * ldout + n] = v;           // ds_store or global_store
      }
    }
  }
}

// ---------------------------------------------------------------------------
// Kernel 2: fused MLP -> shared -> {unit_scores(+noise), cls_logits},
//           top-2, softmax, unit_usage. One block, 256 threads = 8 waves.
// Output layout (floats): probs[128] | top_i[128] | logits[64000] | l2[1] | usage[64]
// ---------------------------------------------------------------------------
__global__ __launch_bounds__(256) void mlp_router_kernel(
    const float* __restrict__ pooled, const float* __restrict__ noise,
    const float* __restrict__ W1, const float* __restrict__ b1,
    const float* __restrict__ W2, const float* __restrict__ b2,
    const float* __restrict__ W3, const float* __restrict__ b3,
    const float* __restrict__ W4, const float* __restrict__ b4,
    const float* __restrict__ Wu, const float* __restrict__ bu,
    const float* __restrict__ Wc, const float* __restrict__ bc,
    float* __restrict__ out)
{
  extern __shared__ float smem[];
  float* buf0 = smem;                 // 64 x LDH
  float* buf1 = smem + BATCH * LDH;   // 64 x LDH
  __shared__ float usage[64];

  // h1 = relu(pooled @ W1^T + b1)   [64,128], K=512
  wmma_layer<512, 4, 8, 128, true,  true >(pooled, CHAN, W1, b1, buf0, LDH);
  __syncthreads();
  // h2 = relu(h1 @ W2^T + b2)       [64,128], K=128
  wmma_layer<128, 4, 8, 128, true,  true >(buf0, LDH, W2, b2, buf1, LDH);
  __syncthreads();
  // h3 = relu(h2 @ W3^T + b3)       [64,64],  K=128
  wmma_layer<128, 4, 4,  64, true,  true >(buf1, LDH, W3, b3, buf0, LDH);
  __syncthreads();
  // shared = h3 @ W4^T + b4         [64,64],  K=64  (no relu)
  wmma_layer< 64, 4, 4,  64, false, true >(buf0, LDH, W4, b4, buf1, LDH);
  __syncthreads();
  // unit_scores = shared @ Wu^T + bu   -> buf0
  wmma_layer< 64, 4, 4,  64, false, true >(buf1, LDH, Wu, bu, buf0, LDH);
  __syncthreads();
  // + NOISE_STD * noise
  for (int i = threadIdx.x; i < BATCH * 64; i += 256) {
    const int bb = i >> 6, uu = i & 63;
    buf0[bb * LDH + uu] += 0.01f * noise[i];
  }
  if (threadIdx.x < 64) usage[threadIdx.x] = 0.0f;
  __syncthreads();
  // cls_logits = shared @ Wc^T + bc -> global [64,1000] (63 N-tiles, padded)
  wmma_layer< 64, 4, 63, 1000, false, false>(buf1, LDH, Wc, bc, out + 256, 1000);
  __syncthreads();

  // top-2 per row, softmax over the 2, usage scatter (LDS ds_add_f32)
  if (threadIdx.x < 64) {
    const int bb = threadIdx.x;
    const float* row = buf0 + bb * LDH;
    float m1 = -INFINITY, m2 = -INFINITY;
    int i1 = 0, i2 = 0;
    for (int u = 0; u < 64; ++u) {
      const float v = row[u];
      if (v > m1)      { m2 = m1; i2 = i1; m1 = v; i1 = u; }
      else if (v > m2) { m2 = v;  i2 = u; }
    }
    const float e  = __expf(m2 - m1);
    const float p1 = 1.0f / (1.0f + e);
    out[2 * bb]           = p1;
    out[2 * bb + 1]       = e * p1;
    out[128 + 2 * bb]     = (float)i1;   // top_i as output dtype
    out[128 + 2 * bb + 1] = (float)i2;
    atomicAdd(&usage[i1], 1.0f);
    atomicAdd(&usage[i2], 1.0f);
  }
  __syncthreads();
  if (threadIdx.x < 64)
    out[64257 + threadIdx.x] = usage[threadIdx.x] * (1.0f / 128.0f); // /(B*TOP_K)
}

// ---------------------------------------------------------------------------
extern "C" void kernel_launch(void* const* d_in, const int* in_sizes, int n_in,
                              void* d_out, int out_size, void* d_ws, size_t ws_size,
                              hipStream_t stream) {
  (void)in_sizes; (void)n_in; (void)out_size; (void)ws_size;
  const float* x     = (const float*)d_in[0];
  const float* noise = (const float*)d_in[1];
  const float* W1 = (const float*)d_in[2];  const float* b1 = (const float*)d_in[3];
  const float* W2 = (const float*)d_in[4];  const float* b2 = (const float*)d_in[5];
  const float* W3 = (const float*)d_in[6];  const float* b3 = (const float*)d_in[7];
  const float* W4 = (const float*)d_in[8];  const float* b4 = (const float*)d_in[9];
  const float* Wu = (const float*)d_in[10]; const float* bu = (const float*)d_in[11];
  const float* Wc = (const float*)d_in[12]; const float* bc = (const float*)d_in[13];
  float* out    = (float*)d_out;
  float* pooled = (float*)d_ws;             // 64*512 floats = 128 KB scratch

  Params12 prm;
  for (int i = 0; i < 12; ++i) prm.p[i] = (const float*)d_in[2 + i];

  // Kernel 1: 4096 pooling blocks + 1 L2-reduction block (runs under HBM time)
  pool_l2_kernel<<<POOL_BLOCKS + 1, 256, 0, stream>>>(x, pooled, out + 64256, prm);

  // Kernel 2: fused MLP/router, 1 block x 256 threads, 2 LDS ping-pong bufs
  const size_t smem = (size_t)2 * BATCH * LDH * sizeof(float);  // 67584 B
  mlp_router_kernel<<<1, 256, smem, stream>>>(pooled, noise,
      W1, b1, W2, b2, W3, b3, W4, b4, Wu, bu, Wc, bc, out);
}